// ChamferDistancePytorch_8254927143420
// MI455X (gfx1250) — compile-verified
//
#include <hip/hip_runtime.h>
#include <hip/hip_bf16.h>

typedef __attribute__((ext_vector_type(2))) float v2f;
typedef __attribute__((ext_vector_type(4))) float v4f;
typedef __attribute__((ext_vector_type(8))) float v8f;

#define N_BATCH 8
#define P1 4096
#define P2 4096
#define DIM 32
#define ROWS_PER_BLOCK 128   // 8 waves x 16 rows
#define JCHUNK 128           // y columns staged in LDS per iteration
#define YPITCH 36            // padded LDS row pitch (dwords): 16B-aligned rows, conflict-free stride

// ---------------------------------------------------------------------------
// Kernel 1: init global column-min workspace to +inf (as int bit pattern).
// ---------------------------------------------------------------------------
__global__ void init_colmin(int* __restrict__ colmin_g, int n)
{
    int i = blockIdx.x * blockDim.x + threadIdx.x;
    if (i < n) colmin_g[i] = 0x7F800000;   // +inf
}

// ---------------------------------------------------------------------------
// Kernel 2: tiled fp32-WMMA distance tiles.
//   block = (batch n, 128-row band of x). 8 waves, each owns 16 rows.
//   Streams y in 128-column LDS chunks; 16x16 tiles via V_WMMA_F32_16X16X4_F32.
//   Hot loop tracks clamped SQUARED distances (min(sqrt(s)) == sqrt(min(s)));
//   sqrt is applied only once per row / column off the hot path.
// ---------------------------------------------------------------------------
__global__ __launch_bounds__(256) void chamfer_tiles(const float* __restrict__ x,
                                                     const float* __restrict__ y,
                                                     float* __restrict__ rowmin_g,
                                                     int*   __restrict__ colmin_g)
{
    __shared__ float ylds[JCHUNK * YPITCH];
    __shared__ float y2s[JCHUNK];
    __shared__ int   colmin[P2];

    const int n    = blockIdx.x / (P1 / ROWS_PER_BLOCK);
    const int ib   = blockIdx.x % (P1 / ROWS_PER_BLOCK);
    const int tid  = threadIdx.x;
    const int wave = tid >> 5;
    const int lane = tid & 31;
    const int half = lane >> 4;    // 0: lanes 0-15 (K={0,1}), 1: lanes 16-31 (K={2,3})
    const int l16  = lane & 15;

    const float* xb = x + (size_t)n * P1 * DIM;
    const float* yb = y + (size_t)n * P2 * DIM;

    // init block-local column minima to +inf
    for (int j = tid; j < P2; j += 256) colmin[j] = 0x7F800000;

    // ---- load this wave's fixed A tile: x[i0..i0+15][0..31], fp32 WMMA A layout ----
    // A 16x4 f32: VGPR0 = K=0 (lanes 0-15) / K=2 (lanes 16-31); VGPR1 = K=1 / K=3.
    const int i0 = ib * ROWS_PER_BLOCK + wave * 16;
    const float* xrow = xb + (size_t)(i0 + l16) * DIM + 2 * half;
    v2f a[8];
    float x2p = 0.0f;
#pragma unroll
    for (int c = 0; c < 8; ++c) {
        a[c] = *(const v2f*)(xrow + 4 * c);
        x2p += a[c].x * a[c].x + a[c].y * a[c].y;
    }
    // each half holds complementary dims -> combine for full |x_i|^2
    float x2 = x2p + __shfl_xor(x2p, 16, 32);
    // C layout: VGPR v, half h -> row M = v + 8*h. Broadcast the right |x|^2 per v.
    float x2row[8];
#pragma unroll
    for (int v = 0; v < 8; ++v)
        x2row[v] = __shfl(x2, v + 8 * half, 32);

    float rowmin[8];   // running min of clamped SQUARED distance per row
#pragma unroll
    for (int v = 0; v < 8; ++v) rowmin[v] = 3.0e38f;

    __syncthreads();

    for (int jbase = 0; jbase < P2; jbase += JCHUNK) {
        // ---- stage y[jbase..jbase+127][0..31] into padded LDS with all 256 threads ----
        // thread pair (2r, 2r+1) covers row r: 16 dims each; |y|^2 joined via shfl_xor(1).
        {
            const int row  = tid >> 1;
            const int hcol = (tid & 1) * 16;
            const float* yrow = yb + (size_t)(jbase + row) * DIM + hcol;
            float s = 0.0f;
#pragma unroll
            for (int q = 0; q < 4; ++q) {
                v4f t = *(const v4f*)(yrow + 4 * q);
                *(v4f*)&ylds[row * YPITCH + hcol + 4 * q] = t;  // 16B-aligned ds_store_b128
                s += t.x * t.x + t.y * t.y + t.z * t.z + t.w * t.w;
            }
            s += __shfl_xor(s, 1, 32);
            if ((tid & 1) == 0) y2s[row] = s;
        }
        __syncthreads();

        // strength-reduced walking pointers (one v_add per tile each)
        const float* bl  = &ylds[l16 * YPITCH + 2 * half];
        const float* y2p = &y2s[l16];
        int*         cmp = &colmin[jbase + l16];

#pragma unroll 2   // two independent WMMA accumulator chains in flight
        for (int jt = 0; jt < JCHUNK / 16; ++jt) {
            // B 4x16 f32 layout mirrors A: lane holds y[j][4c + 2*half .. +1]
            v2f b[8];
#pragma unroll
            for (int c = 0; c < 8; ++c)
                b[c] = *(const v2f*)(bl + 4 * c);

            v8f acc = {};
#pragma unroll
            for (int c = 0; c < 8; ++c)
                acc = __builtin_amdgcn_wmma_f32_16x16x4_f32(
                        /*neg_a=*/false, a[c], /*neg_b=*/false, b[c],
                        /*c_mod=*/(short)0, acc, /*reuse_a=*/false, /*reuse_b=*/false);

            const float y2 = y2p[jt * 16];
            float cmin = 3.0e38f;
#pragma unroll
            for (int v = 0; v < 8; ++v) {
                // clamped squared distance: sqrt deferred (monotonic under min)
                float sq = fmaxf(x2row[v] + y2 - 2.0f * acc[v], 0.0f);
                rowmin[v] = fminf(rowmin[v], sq);
                cmin      = fminf(cmin, sq);
            }
            // lanes L and L+16 hit the same address: the LDS atomic itself merges
            // the two halves -> no cross-lane shuffle, no EXEC masking.
            atomicMin(cmp, __float_as_int(cmin));   // values >= 0: int == float order

            bl  += 16 * YPITCH;
            cmp += 16;
        }
        __syncthreads();   // chunk done before restaging ylds
    }

    // ---- finalize row minima: reduce over the 16 lanes of each half, then sqrt ----
#pragma unroll
    for (int v = 0; v < 8; ++v) {
        float m = rowmin[v];
        m = fminf(m, __shfl_xor(m, 1, 32));
        m = fminf(m, __shfl_xor(m, 2, 32));
        m = fminf(m, __shfl_xor(m, 4, 32));
        m = fminf(m, __shfl_xor(m, 8, 32));
        rowmin[v] = __builtin_sqrtf(m);   // only 8 sqrts per wave, off the hot path
    }
    if (l16 == 0) {  // lane 0 -> rows i0..i0+7, lane 16 -> rows i0+8..i0+15
#pragma unroll
        for (int v = 0; v < 8; ++v)
            rowmin_g[(size_t)n * P1 + i0 + 8 * half + v] = rowmin[v];
    }

    __syncthreads();
    // ---- merge block-local column minima (squared) into global (32 atomics/address) ----
    for (int j = tid; j < P2; j += 256)
        atomicMin(&colmin_g[(size_t)n * P2 + j], colmin[j]);
}

// ---------------------------------------------------------------------------
// Kernel 3: per-batch means -> output (N floats). colmin holds squared minima.
// ---------------------------------------------------------------------------
__global__ __launch_bounds__(256) void chamfer_reduce(const float* __restrict__ rowmin_g,
                                                      const int*   __restrict__ colmin_g,
                                                      float* __restrict__ out)
{
    __shared__ float s1[256];
    __shared__ float s2[256];
    const int n = blockIdx.x, tid = threadIdx.x;
    float sa = 0.0f, sb = 0.0f;
    for (int i = tid; i < P1; i += 256) sa += rowmin_g[(size_t)n * P1 + i];
    for (int j = tid; j < P2; j += 256)
        sb += __builtin_sqrtf(__int_as_float(colmin_g[(size_t)n * P2 + j]));
    s1[tid] = sa; s2[tid] = sb;
    __syncthreads();
    for (int st = 128; st > 0; st >>= 1) {
        if (tid < st) { s1[tid] += s1[tid + st]; s2[tid] += s2[tid + st]; }
        __syncthreads();
    }
    if (tid == 0)
        out[n] = 0.5f * (s1[0] / (float)P1 + s2[0] / (float)P2);
}

extern "C" void kernel_launch(void* const* d_in, const int* in_sizes, int n_in,
                              void* d_out, int out_size, void* d_ws, size_t ws_size,
                              hipStream_t stream)
{
    const float* x = (const float*)d_in[0];   // (8, 4096, 32) fp32
    const float* y = (const float*)d_in[1];   // (8, 4096, 32) fp32
    float* out = (float*)d_out;               // (8,) fp32

    float* rowmin_g = (float*)d_ws;                                   // N*P1 floats
    int*   colmin_g = (int*)((char*)d_ws + (size_t)N_BATCH * P1 * 4); // N*P2 ints

    init_colmin<<<(N_BATCH * P2 + 255) / 256, 256, 0, stream>>>(colmin_g, N_BATCH * P2);
    chamfer_tiles<<<N_BATCH * (P1 / ROWS_PER_BLOCK), 256, 0, stream>>>(x, y, rowmin_g, colmin_g);
    chamfer_reduce<<<N_BATCH, 256, 0, stream>>>(rowmin_g, colmin_g, out);
}